// RobustSigmaDistance_27204322853025
// MI455X (gfx1250) — compile-verified
//
#include <hip/hip_runtime.h>
#include <math.h>

typedef __attribute__((ext_vector_type(2))) float v2f;
typedef __attribute__((ext_vector_type(8))) float v8f;

#define NPTS  4096
#define BATCH 4
#define NV    (NPTS * 3)      // 12288 residual scalars per (batch, direction)
#define NS    16384           // sort size (power of two, padded with +INF)

// ---------------------------------------------------------------------------
// Stage 1: bidirectional nearest-neighbor residuals via V_WMMA_F32_16X16X4_F32
// score(i,j) = q_i . c_j - 0.5*|c_j|^2 ; argmax_j score == argmin_j d^2
// One wave handles 16 query points; candidates staged in LDS as float4.
// Candidate float4 is prefetched one chunk ahead (register double buffer) so
// the ds_load_b128 latency overlaps the WMMA + compare/select of the current
// chunk; B-tile half selection is branchless (v_cndmask, uniform EXEC).
// ---------------------------------------------------------------------------
__global__ __launch_bounds__(256) void nn_residual_kernel(
    const float* __restrict__ x, const float* __restrict__ y,
    float* __restrict__ resid) {
  extern __shared__ float4 cand[];  // 4096 * 16B = 64KB dynamic LDS

  const int blocksPerBD = NPTS / 128;            // 32 blocks per (batch,dir)
  const int bd     = blockIdx.x / blocksPerBD;   // 0..7
  const int qchunk = blockIdx.x % blocksPerBD;
  const int b = bd >> 1, dir = bd & 1;
  const float* S1 = (dir == 0) ? (x + (size_t)b * NPTS * 3) : (y + (size_t)b * NPTS * 3);
  const float* S2 = (dir == 0) ? (y + (size_t)b * NPTS * 3) : (x + (size_t)b * NPTS * 3);

  // Stage candidate coords + half squared norm into LDS.
  for (int i = threadIdx.x; i < NPTS; i += 256) {
    float cx = S2[i * 3 + 0], cy = S2[i * 3 + 1], cz = S2[i * 3 + 2];
    cand[i] = make_float4(cx, cy, cz, 0.5f * (cx * cx + cy * cy + cz * cz));
  }
  __syncthreads();

  const int wave = threadIdx.x >> 5;
  const int lane = threadIdx.x & 31;
  const bool hi  = (lane & 16) != 0;  // half 0: K dims {0,1}; half 1: {2,3}
  const int l16  = lane & 15;
  const int qbase = qchunk * 128 + wave * 16;

  // A-matrix tile: 16 queries x 4 dims (dim 3 zero-padded). Branchless select.
  const int m = qbase + l16;
  float a0 = S1[m * 3 + 0], a1 = S1[m * 3 + 1], a2 = S1[m * 3 + 2];
  v2f a;
  a.x = hi ? a2 : a0;
  a.y = hi ? 0.0f : a1;

  float best[8];
  int   bidx[8];
#pragma unroll
  for (int r = 0; r < 8; ++r) { best[r] = -INFINITY; bidx[r] = 0x7fffffff; }

  float4 cd = cand[l16];  // prefetch chunk 0
#pragma unroll 2
  for (int ch = 0; ch < NPTS / 16; ++ch) {
    const int c = ch * 16 + l16;              // candidate column for this lane
    float4 nx = cand[(c + 16) & (NPTS - 1)];  // prefetch next chunk (wraps)
    v2f bm;
    bm.x = hi ? cd.z : cd.x;                  // branchless B tile
    bm.y = hi ? 0.0f : cd.y;
    const float hn = cd.w;
    v8f acc = {0.f, 0.f, 0.f, 0.f, 0.f, 0.f, 0.f, 0.f};
    v8f d = __builtin_amdgcn_wmma_f32_16x16x4_f32(
        /*neg_a=*/false, a, /*neg_b=*/false, bm,
        /*c_mod=*/(short)0, acc, /*reuse_a=*/false, /*reuse_b=*/false);
#pragma unroll
    for (int r = 0; r < 8; ++r) {
      float s = d[r] - hn;
      if (s > best[r]) { best[r] = s; bidx[r] = c; }  // ascending c: first max kept
    }
    cd = nx;
  }

  // Reduce each row's (best, idx) across the 16 lanes of its half-wave.
#pragma unroll
  for (int r = 0; r < 8; ++r) {
    float bs = best[r]; int bi = bidx[r];
#pragma unroll
    for (int msk = 1; msk < 16; msk <<= 1) {
      float os = __shfl_xor(bs, msk, 32);
      int   oi = __shfl_xor(bi, msk, 32);
      if (os > bs || (os == bs && oi < bi)) { bs = os; bi = oi; }
    }
    best[r] = bs; bidx[r] = bi;
  }

  // Lanes 0 and 16 write residuals for rows qbase+0..7 and qbase+8..15.
  if (l16 == 0) {
    const int rowoff = hi ? 8 : 0;
#pragma unroll
    for (int r = 0; r < 8; ++r) {
      int row = qbase + rowoff + r;
      float4 cj = cand[bidx[r]];
      float* o = resid + (size_t)bd * NV + (size_t)row * 3;
      o[0] = S1[row * 3 + 0] - cj.x;
      o[1] = S1[row * 3 + 1] - cj.y;
      o[2] = S1[row * 3 + 2] - cj.z;
    }
  }
}

// ---------------------------------------------------------------------------
// Stage 2: per (batch,dir): bitonic sort in LDS -> quantiles -> masked stds.
// ---------------------------------------------------------------------------
__global__ __launch_bounds__(1024) void robust_std_kernel(
    const float* __restrict__ resid, float* __restrict__ stats) {
  extern __shared__ float sbuf[];  // NS floats = 64KB dynamic LDS
  __shared__ double redA[32], redB[32];
  __shared__ int    redCA[32], redCB[32];
  __shared__ float  qv[4];
  __shared__ float  s_meanB, s_meanE;
  __shared__ int    s_cntB, s_cntE;

  const int bd  = blockIdx.x;
  const int tid = threadIdx.x;
  const float* v = resid + (size_t)bd * NV;

  for (int i = tid; i < NS; i += 1024) sbuf[i] = (i < NV) ? v[i] : INFINITY;
  __syncthreads();

  // Bitonic sort (ascending).
  for (int k = 2; k <= NS; k <<= 1) {
    for (int j = k >> 1; j > 0; j >>= 1) {
      for (int i = tid; i < NS; i += 1024) {
        int ixj = i ^ j;
        if (ixj > i) {
          float av = sbuf[i], bv = sbuf[ixj];
          bool up = ((i & k) == 0);
          if ((av > bv) == up) { sbuf[i] = bv; sbuf[ixj] = av; }
        }
      }
      __syncthreads();
    }
  }

  // Linear-interpolated quantiles at [0.05, 0.95, 0.25, 0.75].
  if (tid == 0) {
    const float Q[4] = {0.05f, 0.95f, 0.25f, 0.75f};
    for (int qi = 0; qi < 4; ++qi) {
      float h = Q[qi] * (float)(NV - 1);
      int lo = (int)floorf(h);
      float fr = h - (float)lo;
      qv[qi] = sbuf[lo] + fr * (sbuf[lo + 1] - sbuf[lo]);
    }
  }
  __syncthreads();
  const float q0 = qv[0], q1 = qv[1], q2 = qv[2], q3 = qv[3];

  // Pass 1: masked counts + sums (double accumulation).
  double sB = 0.0, sE = 0.0; int cB = 0, cE = 0;
  for (int i = tid; i < NV; i += 1024) {
    float val = sbuf[i];
    if (val < q0 || val > q1) { sB += (double)val; ++cB; }
    if (val > q2 && val < q3) { sE += (double)val; ++cE; }
  }
  for (int msk = 1; msk < 32; msk <<= 1) {
    sB += __shfl_xor(sB, msk, 32);  sE += __shfl_xor(sE, msk, 32);
    cB += __shfl_xor(cB, msk, 32);  cE += __shfl_xor(cE, msk, 32);
  }
  if ((tid & 31) == 0) { int w = tid >> 5; redA[w] = sB; redB[w] = sE; redCA[w] = cB; redCB[w] = cE; }
  __syncthreads();
  if (tid < 32) {
    double ra = redA[tid], rb = redB[tid]; int ca = redCA[tid], cb = redCB[tid];
    for (int msk = 1; msk < 32; msk <<= 1) {
      ra += __shfl_xor(ra, msk, 32); rb += __shfl_xor(rb, msk, 32);
      ca += __shfl_xor(ca, msk, 32); cb += __shfl_xor(cb, msk, 32);
    }
    if (tid == 0) {
      s_meanB = (float)(ra / (double)ca);
      s_meanE = (float)(rb / (double)cb);
      s_cntB = ca; s_cntE = cb;
    }
  }
  __syncthreads();
  const float mB = s_meanB, mE = s_meanE;

  // Pass 2: masked sums of squared deviations.
  double dB = 0.0, dE = 0.0;
  for (int i = tid; i < NV; i += 1024) {
    float val = sbuf[i];
    if (val < q0 || val > q1) { double t = (double)(val - mB); dB += t * t; }
    if (val > q2 && val < q3) { double t = (double)(val - mE); dE += t * t; }
  }
  for (int msk = 1; msk < 32; msk <<= 1) {
    dB += __shfl_xor(dB, msk, 32);  dE += __shfl_xor(dE, msk, 32);
  }
  __syncthreads();
  if ((tid & 31) == 0) { int w = tid >> 5; redA[w] = dB; redB[w] = dE; }
  __syncthreads();
  if (tid < 32) {
    double ra = redA[tid], rb = redB[tid];
    for (int msk = 1; msk < 32; msk <<= 1) {
      ra += __shfl_xor(ra, msk, 32); rb += __shfl_xor(rb, msk, 32);
    }
    if (tid == 0) {
      stats[bd * 2 + 0] = (float)sqrt(ra / (double)(s_cntB - 1));
      stats[bd * 2 + 1] = (float)sqrt(rb / (double)(s_cntE - 1));
    }
  }
}

// ---------------------------------------------------------------------------
// Stage 3: max over directions per batch, mean over batches -> 2 outputs.
// ---------------------------------------------------------------------------
__global__ void finalize_kernel(const float* __restrict__ stats,
                                float* __restrict__ out) {
  if (blockIdx.x == 0 && threadIdx.x == 0) {
    float sb = 0.f, se = 0.f;
    for (int b = 0; b < BATCH; ++b) {
      sb += fmaxf(stats[(b * 2 + 0) * 2 + 0], stats[(b * 2 + 1) * 2 + 0]);
      se += fmaxf(stats[(b * 2 + 0) * 2 + 1], stats[(b * 2 + 1) * 2 + 1]);
    }
    out[0] = sb / (float)BATCH;
    out[1] = se / (float)BATCH;
  }
}

extern "C" void kernel_launch(void* const* d_in, const int* in_sizes, int n_in,
                              void* d_out, int out_size, void* d_ws, size_t ws_size,
                              hipStream_t stream) {
  const float* x = (const float*)d_in[0];   // [4, 4096, 3] f32
  const float* y = (const float*)d_in[1];   // [4, 4096, 3] f32
  float* resid = (float*)d_ws;              // 8 * 12288 floats = 384KB
  float* stats = resid + 8 * NV;            // 16 floats
  float* out   = (float*)d_out;             // 2 floats

  nn_residual_kernel<<<8 * (NPTS / 128), 256, 64 * 1024, stream>>>(x, y, resid);
  robust_std_kernel<<<8, 1024, NS * sizeof(float), stream>>>(resid, stats);
  finalize_kernel<<<1, 32, 0, stream>>>(stats, out);
}